// PostProcessor_60455959659013
// MI455X (gfx1250) — compile-verified
//
#include <hip/hip_runtime.h>
#include <hip/hip_bf16.h>

#define TPB 256

// ---------------------------------------------------------------------------
// Rotated 3D box IoU (BEV convex intersection x height overlap), one thread
// per box pair.  Box layout per row (7 floats): [x, y, z, dx, h, dz, ry].
// BEV uses (x, z, dx, dz, ry); height uses (y, h) with the reference's
// "min_h = -(y + h/2), max_h = -(y - h/2)" convention and the quirk that
// area_a is scaled by h_a0 = height of pred row 0 for EVERY element.
// ---------------------------------------------------------------------------

__device__ __forceinline__ void make_corners(float x, float z, float dx, float dz,
                                             float ry, float* cx, float* cz) {
    float s, c;
    sincosf(ry, &s, &c);
    const float hx = dx * 0.5f, hz = dz * 0.5f;
    const float lx[4] = { hx, -hx, -hx,  hx };
    const float lz[4] = { hz,  hz, -hz, -hz };
#pragma unroll
    for (int k = 0; k < 4; ++k) {
        cx[k] = lx[k] * c - lz[k] * s + x;
        cz[k] = lx[k] * s + lz[k] * c + z;
    }
}

__device__ __forceinline__ float quad_area(const float* cx, const float* cz) {
    float a = 0.f;
#pragma unroll
    for (int k = 0; k < 4; ++k) {
        const int k2 = (k + 1) & 3;
        a += cx[k] * cz[k2] - cz[k] * cx[k2];
    }
    return 0.5f * fabsf(a);
}

// Sutherland-Hodgman: clip CCW quad A by CCW quad B; return intersection area.
__device__ float convex_intersect_area(const float* ax, const float* az,
                                       const float* bx, const float* bz) {
    float px[10], pz[10], qx[10], qz[10], d[10];
    int n = 4;
#pragma unroll
    for (int k = 0; k < 4; ++k) { px[k] = ax[k]; pz[k] = az[k]; }

#pragma unroll
    for (int e = 0; e < 4; ++e) {
        const int   e2  = (e + 1) & 3;
        const float ex  = bx[e2] - bx[e];
        const float ez  = bz[e2] - bz[e];
        const float bxe = bx[e],  bze = bz[e];

        for (int k = 0; k < n; ++k)
            d[k] = ex * (pz[k] - bze) - ez * (px[k] - bxe);

        int m = 0;
        for (int k = 0; k < n; ++k) {
            const int   k2  = (k + 1 == n) ? 0 : k + 1;
            const float dk  = d[k];
            const float dk2 = d[k2];
            const bool  ink  = (dk  >= 0.f);
            const bool  ink2 = (dk2 >= 0.f);
            if (ink && m < 10) { qx[m] = px[k]; qz[m] = pz[k]; ++m; }
            if (ink != ink2 && m < 10) {
                const float t = dk / (dk - dk2);
                qx[m] = px[k] + t * (px[k2] - px[k]);
                qz[m] = pz[k] + t * (pz[k2] - pz[k]);
                ++m;
            }
        }
        n = m;
        for (int k = 0; k < n; ++k) { px[k] = qx[k]; pz[k] = qz[k]; }
        if (n == 0) break;
    }

    if (n < 3) return 0.f;
    float a = 0.f;
    for (int k = 0; k < n; ++k) {
        const int k2 = (k + 1 == n) ? 0 : k + 1;
        a += px[k] * pz[k2] - pz[k] * px[k2];
    }
    return 0.5f * fabsf(a);
}

__global__ __launch_bounds__(TPB) void iou3d_kernel(const float* __restrict__ pred,
                                                    const float* __restrict__ targ,
                                                    float* __restrict__ out, int n) {
    __shared__ float sp[TPB * 7];   // 7168 B, 16B-aligned block tiles
    __shared__ float st[TPB * 7];

    const int tid    = threadIdx.x;
    const int block0 = blockIdx.x * TPB;

    // ---- coalesced async global->LDS staging (CDNA5 ASYNCcnt path) ----
    int boxes = n - block0;
    if (boxes > TPB) boxes = TPB;
    const int nbytes   = boxes * 28;
    const int nvec     = nbytes >> 4;          // full 16-byte vectors
    const int ndw_tail = (nbytes & 15) >> 2;   // leftover dwords

    const unsigned lp = (unsigned)(uintptr_t)&sp[0];   // addrspace(3) offset
    const unsigned lt = (unsigned)(uintptr_t)&st[0];
    const float* gp = pred + (size_t)block0 * 7;
    const float* gt = targ + (size_t)block0 * 7;

    for (int j = tid; j < nvec; j += TPB) {
        const unsigned off = (unsigned)j << 4;
        const unsigned lap = lp + off;
        const unsigned lat = lt + off;
        asm volatile("global_load_async_to_lds_b128 %0, %1, %2 offset:0"
                     :: "v"(lap), "v"(off), "s"(gp) : "memory");
        asm volatile("global_load_async_to_lds_b128 %0, %1, %2 offset:0"
                     :: "v"(lat), "v"(off), "s"(gt) : "memory");
    }
    for (int j = tid; j < ndw_tail; j += TPB) {
        const unsigned off = ((unsigned)nvec << 4) + ((unsigned)j << 2);
        const unsigned lap = lp + off;
        const unsigned lat = lt + off;
        asm volatile("global_load_async_to_lds_b32 %0, %1, %2 offset:0"
                     :: "v"(lap), "v"(off), "s"(gp) : "memory");
        asm volatile("global_load_async_to_lds_b32 %0, %1, %2 offset:0"
                     :: "v"(lat), "v"(off), "s"(gt) : "memory");
    }
    asm volatile("s_wait_asynccnt 0x0" ::: "memory");
    __syncthreads();

    const int gi = block0 + tid;
    if (gi >= n) return;

    float p[7], q[7];
#pragma unroll
    for (int k = 0; k < 7; ++k) { p[k] = sp[tid * 7 + k]; q[k] = st[tid * 7 + k]; }

    // reference quirk: h_a0 from pred row 0, with identical rounding order
    const float y0  = pred[1];
    const float hh0 = pred[4];
    const float h_a0 = (-(y0 - hh0 * 0.5f)) - (-(y0 + hh0 * 0.5f));

    // height overlap
    const float min_h_a = -(p[1] + p[4] * 0.5f);
    const float max_h_a = -(p[1] - p[4] * 0.5f);
    const float min_h_b = -(q[1] + q[4] * 0.5f);
    const float max_h_b = -(q[1] - q[4] * 0.5f);
    float h_ov = fminf(max_h_a, max_h_b) - fmaxf(min_h_a, min_h_b);
    h_ov = fmaxf(h_ov, 0.f);

    // BEV corners (CCW)
    float ax[4], az[4], bx[4], bz[4];
    make_corners(p[0], p[2], p[3], p[5], p[6], ax, az);
    make_corners(q[0], q[2], q[3], q[5], q[6], bx, bz);

    const float overlap   = convex_intersect_area(ax, az, bx, bz);
    const float overlap3d = overlap * h_ov;
    const float area_a    = quad_area(ax, az);
    const float area_b    = quad_area(bx, bz);
    const float t_h       = max_h_b - min_h_b;
    const float union3d   = area_a * h_a0 + area_b * t_h - overlap3d;

    out[gi] = overlap3d / union3d;
}

extern "C" void kernel_launch(void* const* d_in, const int* in_sizes, int n_in,
                              void* d_out, int out_size, void* d_ws, size_t ws_size,
                              hipStream_t stream) {
    (void)n_in; (void)out_size; (void)d_ws; (void)ws_size;
    const float* pred = (const float*)d_in[0];
    const float* targ = (const float*)d_in[1];
    float* out = (float*)d_out;
    const int n = in_sizes[0] / 7;
    const int blocks = (n + TPB - 1) / TPB;
    iou3d_kernel<<<blocks, TPB, 0, stream>>>(pred, targ, out, n);
}